// AlternateLayer_3023656976552
// MI455X (gfx1250) — compile-verified
//
#include <hip/hip_runtime.h>
#include <math.h>

#define BB 32
#define SS 128
#define DD 15000
#define NG 30          // time-distributed groups per (b,s)
#define GL 500         // group length
#define HN (BB*NG*SS)  // 122880 h elements
#define NSTEP 30
#define NH 512         // 4*S lstm gate width

typedef __attribute__((ext_vector_type(2))) float v2f;
typedef __attribute__((ext_vector_type(4))) float v4f;
typedef __attribute__((ext_vector_type(8))) float v8f;

__device__ __forceinline__ float sigf(float x) { return 1.0f / (1.0f + __expf(-x)); }

__device__ __forceinline__ float wredsum(float v) {
  #pragma unroll
  for (int off = 16; off; off >>= 1) v += __shfl_xor(v, off, 32);
  return v;
}

// D(16x16,f32) += A(16x4,f32) x B(4x16,f32)  -> v_wmma_f32_16x16x4_f32
__device__ __forceinline__ v8f wmma4(v2f a, v2f b, v8f c) {
  return __builtin_amdgcn_wmma_f32_16x16x4_f32(false, a, false, b, (short)0, c,
                                               false, false);
}

// ---------------------------------------------------------------------------
// Kernel A: time-distributed gates. Bandwidth-bound: streams all of x once.
// Flip folded into reversed weights; slice j' = 29-j is contiguous in x.
// One wave per 500-float group, float4 coalesced loads, shuffle reduce.
// ---------------------------------------------------------------------------
__global__ __launch_bounds__(256) void k_tdist(const float* __restrict__ x,
                                               const float* __restrict__ tdW,
                                               const float* __restrict__ tdb,
                                               float* __restrict__ h) {
  __shared__ float w[4 * GL];  // reversed weights, [c][t]
  for (int i = threadIdx.x; i < 4 * GL; i += 256) {
    int c = i / GL, t = i % GL;
    w[i] = tdW[c * GL + (GL - 1 - t)];
  }
  __syncthreads();
  const int wave = threadIdx.x >> 5;
  const int lane = threadIdx.x & 31;
  const float b0 = tdb[0], b2 = tdb[2], b3 = tdb[3];
  for (int g = blockIdx.x * 8 + wave; g < HN; g += gridDim.x * 8) {
    int jp = g % NG;               // slice index (contiguous in x)
    int s  = (g / NG) % SS;
    int b  = g / (NG * SS);
    const float* px = x + (size_t)(b * SS + s) * DD + jp * GL;
    float a0 = 0.f, a2 = 0.f, a3 = 0.f;  // gates 0,2,3 (gate 1 unused)
    #pragma unroll
    for (int it = 0; it < 4; ++it) {
      int t0 = 4 * lane + 128 * it;
      if (t0 < GL) {
        v4f xv = *(const v4f*)(px + t0);
        v4f w0 = *(const v4f*)(&w[0 * GL + t0]);
        v4f w2 = *(const v4f*)(&w[2 * GL + t0]);
        v4f w3 = *(const v4f*)(&w[3 * GL + t0]);
        a0 += xv.x * w0.x + xv.y * w0.y + xv.z * w0.z + xv.w * w0.w;
        a2 += xv.x * w2.x + xv.y * w2.y + xv.z * w2.z + xv.w * w2.w;
        a3 += xv.x * w3.x + xv.y * w3.y + xv.z * w3.z + xv.w * w3.w;
      }
    }
    a0 = wredsum(a0); a2 = wredsum(a2); a3 = wredsum(a3);
    if (lane == 0) {
      float gi = a0 + b0, gg = a2 + b2, go = a3 + b3;
      float hv = sigf(go) * tanhf(sigf(gi) * tanhf(gg));
      int j = (NG - 1) - jp;       // output group index (flip)
      h[(size_t)b * (NG * SS) + j * SS + s] = hv;
    }
  }
}

// ---------------------------------------------------------------------------
// Kernel B: attention softmax + scale, one block per batch element.
// ---------------------------------------------------------------------------
__global__ __launch_bounds__(256) void k_att(float* __restrict__ h,
                                             const float* __restrict__ attW,
                                             const float* __restrict__ attb) {
  __shared__ float sh[NG * SS];  // 3840
  __shared__ float slog[NG];
  __shared__ float satt[NG];
  const int b = blockIdx.x;
  for (int i = threadIdx.x; i < NG * SS; i += 256)
    sh[i] = h[(size_t)b * NG * SS + i];
  __syncthreads();
  const int wave = threadIdx.x >> 5, lane = threadIdx.x & 31;
  for (int r = wave; r < NG; r += 8) {
    const float* wr = attW + (size_t)r * (NG * SS);
    float p = 0.f;
    for (int i = lane; i < NG * SS; i += 32) p += sh[i] * wr[i];
    p = wredsum(p);
    if (lane == 0) slog[r] = p + attb[r];
  }
  __syncthreads();
  if (threadIdx.x == 0) {
    float m = slog[0];
    for (int r = 1; r < NG; ++r) m = fmaxf(m, slog[r]);
    float ssum = 0.f;
    for (int r = 0; r < NG; ++r) { float e = __expf(slog[r] - m); satt[r] = e; ssum += e; }
    float inv = 1.0f / ssum;
    for (int r = 0; r < NG; ++r) satt[r] *= inv;
  }
  __syncthreads();
  for (int i = threadIdx.x; i < NG * SS; i += 256)
    h[(size_t)b * NG * SS + i] = sh[i] * satt[i >> 7];
}

// ---------------------------------------------------------------------------
// Kernel C: xpart[r] = xt_r @ Wih^T for all 30 steps (parallel), via
// v_wmma_f32_16x16x4_f32. One wave per 16x16 output tile, K=128 in 32 chunks.
// ---------------------------------------------------------------------------
__global__ __launch_bounds__(256) void k_xih(const float* __restrict__ hw,
                                             const float* __restrict__ Wih,
                                             float* __restrict__ xpart) {
  const int tid  = blockIdx.x * 8 + (threadIdx.x >> 5);  // tile id 0..1919
  const int lane = threadIdx.x & 31;
  const int ln   = lane & 15;
  const int hi   = lane >> 4;
  const int khi  = hi * 2;
  const int nt = tid & 31;         // 32 n-tiles  (NH/16)
  const int mt = (tid >> 5) & 1;   // 2  m-tiles  (BB/16)
  const int r  = tid >> 6;         // step 0..29
  v8f acc = {};
  const float* pa = hw + (size_t)(mt * 16 + ln) * (NG * SS) + r * SS;  // xt row
  const float* pb = Wih + (size_t)(nt * 16 + ln) * SS;                 // Wih^T col
  #pragma unroll
  for (int kc = 0; kc < 32; ++kc) {
    v2f a  = *(const v2f*)(pa + 4 * kc + khi);
    v2f bv = *(const v2f*)(pb + 4 * kc + khi);
    acc = wmma4(a, bv, acc);
  }
  #pragma unroll
  for (int i = 0; i < 8; ++i) {
    int M   = mt * 16 + i + 8 * hi;
    int col = nt * 16 + ln;
    xpart[((size_t)r * BB + M) * NH + col] = acc[i];
  }
}

// ---------------------------------------------------------------------------
// Kernel D: sequential LSTM scan + fused output projection. Single 1024-thread
// workgroup; h/c state and the 32x512 gate buffer live in LDS (96 KB of the
// WGP's 320 KB). Recurrent GEMM h @ Whh^T via wmma (A from LDS ds_load_b64).
// ---------------------------------------------------------------------------
__global__ __launch_bounds__(1024) void k_scan(const float* __restrict__ xpart,
                                               const float* __restrict__ Whh,
                                               const float* __restrict__ lstmb,
                                               const float* __restrict__ fdW,
                                               const float* __restrict__ fdb,
                                               float* __restrict__ out) {
  __shared__ float hS[BB * SS];   // 16 KB
  __shared__ float cS[BB * SS];   // 16 KB
  __shared__ float gt[BB * NH];   // 64 KB
  const int tid  = threadIdx.x;
  const int wave = tid >> 5, lane = tid & 31;
  const int ln = lane & 15, hi = lane >> 4, khi = hi * 2;
  for (int i = tid; i < BB * SS; i += 1024) { hS[i] = 0.f; cS[i] = 0.f; }
  __syncthreads();
  for (int r = 0; r < NSTEP; ++r) {
    // phase 1: gt = hS @ Whh^T + xpart[r] + b   (64 tiles over 32 waves)
    #pragma unroll
    for (int tt = 0; tt < 2; ++tt) {
      int t  = wave + tt * 32;
      int nt = t & 31, mt = t >> 5;
      v8f acc = {};
      const float* pa = &hS[(mt * 16 + ln) * SS];
      const float* pb = Whh + (size_t)(nt * 16 + ln) * SS;
      #pragma unroll
      for (int kc = 0; kc < 32; ++kc) {
        v2f a  = *(const v2f*)(pa + 4 * kc + khi);
        v2f bv = *(const v2f*)(pb + 4 * kc + khi);
        acc = wmma4(a, bv, acc);
      }
      #pragma unroll
      for (int i = 0; i < 8; ++i) {
        int M   = mt * 16 + i + 8 * hi;
        int col = nt * 16 + ln;
        gt[M * NH + col] =
            acc[i] + xpart[((size_t)r * BB + M) * NH + col] + lstmb[col];
      }
    }
    __syncthreads();
    // phase 2: elementwise LSTM cell
    for (int i = tid; i < BB * SS; i += 1024) {
      int bb = i >> 7, ss = i & 127;
      const float* g = &gt[bb * NH];
      float ig = g[ss], fg = g[SS + ss], gg = g[2 * SS + ss], og = g[3 * SS + ss];
      float c  = sigf(fg) * cS[i] + sigf(ig) * tanhf(gg);
      float hv = sigf(og) * tanhf(c);
      cS[i] = c;
      hS[i] = hv;
    }
    __syncthreads();
    // phase 3: fused output projection y[b, 2r+q] = tanh(h . fd_W[q] + fd_b[q])
    {
      int b = wave;  // 32 waves == 32 batch rows
      #pragma unroll
      for (int q = 0; q < 2; ++q) {
        v4f hv = *(const v4f*)&hS[b * SS + lane * 4];
        v4f wv = *(const v4f*)(fdW + q * SS + lane * 4);
        float p = hv.x * wv.x + hv.y * wv.y + hv.z * wv.z + hv.w * wv.w;
        p = wredsum(p);
        if (lane == 0) out[b * 60 + r * 2 + q] = tanhf(p + fdb[q]);
      }
    }
    __syncthreads();
  }
}

extern "C" void kernel_launch(void* const* d_in, const int* in_sizes, int n_in,
                              void* d_out, int out_size, void* d_ws,
                              size_t ws_size, hipStream_t stream) {
  const float* x    = (const float*)d_in[0];
  const float* tdW  = (const float*)d_in[1];
  const float* tdb  = (const float*)d_in[2];
  const float* attW = (const float*)d_in[3];
  const float* attb = (const float*)d_in[4];
  const float* Wih  = (const float*)d_in[5];
  const float* Whh  = (const float*)d_in[6];
  const float* lb   = (const float*)d_in[7];
  const float* fdW  = (const float*)d_in[8];
  const float* fdb  = (const float*)d_in[9];
  float* out = (float*)d_out;

  float* h     = (float*)d_ws;               // 122880 f32
  float* xpart = h + (size_t)BB * NG * SS;   // 491520 f32

  k_tdist<<<1920, 256, 0, stream>>>(x, tdW, tdb, h);
  k_att  <<<32,   256, 0, stream>>>(h, attW, attb);
  k_xih  <<<240,  256, 0, stream>>>(h, Wih, xpart);
  k_scan <<<1,   1024, 0, stream>>>(xpart, Whh, lb, fdW, fdb, out);
}